// Quantumnet_28432683499990
// MI455X (gfx1250) — compile-verified
//
#include <hip/hip_runtime.h>
#include <math.h>

typedef float v2f __attribute__((ext_vector_type(2)));
typedef float v8f __attribute__((ext_vector_type(8)));

#define KDIM 512
#define LDS_STRIDE (KDIM + 2)   // 514 dwords row stride -> no LDS bank conflicts across N rows
#define NOUT 10

__global__ __launch_bounds__(256) void qnet_fused_kernel(
    const float* __restrict__ X,      // (B, 512)
    const float* __restrict__ preW,   // (3, 512)
    const float* __restrict__ preB,   // (3,)
    const float* __restrict__ qp,     // (45,) = (15, 3)
    const float* __restrict__ postW,  // (10, 3)
    const float* __restrict__ postB,  // (10,)
    float* __restrict__ out,          // (B, 10)
    int batch)
{
    // Padded B tile: 16 N-rows x 512 K, rows 3..15 are zero (N padding for WMMA).
    __shared__ float sB[16 * LDS_STRIDE];
    // Per-wave scatter of the WMMA accumulator columns 0..2 (16B-aligned rows).
    __shared__ float sC[8][16][4];

    const int tid  = threadIdx.x;
    const int wave = tid >> 5;
    const int lane = tid & 31;

    // ---- stage pre_W into padded/zeroed LDS (once per block) ----
    for (int i = tid; i < 16 * LDS_STRIDE; i += 256) {
        int n = i / LDS_STRIDE;
        int k = i - n * LDS_STRIDE;
        float v = 0.0f;
        if (n < 3 && k < KDIM) v = preW[n * KDIM + k];
        sB[i] = v;
    }
    __syncthreads();

    const int rowBase = (blockIdx.x * 8 + wave) * 16;   // wave-uniform
    const bool waveActive = rowBase < batch;            // wave-uniform -> EXEC stays all-1s inside

    if (waveActive) {
        // ---- WMMA f32 16x16x4 accumulation over K = 512 ----
        // A layout: lanes 0-15 -> row m=lane, K = {k0, k0+1}
        //           lanes 16-31 -> row m=lane-16, K = {k0+2, k0+3}
        // B layout (mirror): lanes 0-15 -> col n=lane, K = {k0, k0+1}; lanes 16-31 -> K+2
        const int m    = lane & 15;
        const int koff = (lane >> 4) * 2;

        int mrow = rowBase + m;
        if (mrow >= batch) mrow = batch - 1;            // clamp partial tile (no OOB reads)

        const float* aPtr = X  + (size_t)mrow * KDIM + koff;
        const float* bPtr = sB + m * LDS_STRIDE + koff;

        v8f acc = {};
        #pragma unroll
        for (int k0 = 0; k0 < KDIM; k0 += 4) {
            if ((k0 & 63) == 0)                          // stream-prefetch 1KB ahead
                __builtin_prefetch(aPtr + k0 + 256, 0, 0);
            v2f a = *(const v2f*)(aPtr + k0);
            v2f b = *(const v2f*)(bPtr + k0);
            acc = __builtin_amdgcn_wmma_f32_16x16x4_f32(
                false, a, false, b, (short)0, acc, false, false);
        }

        // ---- unpick C/D layout: VGPR v, lanes 0-15 -> (M=v,   N=lane),
        //                                  lanes 16-31 -> (M=v+8, N=lane-16) ----
        const int col  = lane & 15;
        const int rOff = (lane >> 4) * 8;
        if (col < 3) {
            #pragma unroll
            for (int v = 0; v < 8; ++v)
                sC[wave][rOff + v][col] = acc[v];
        }
    }
    __syncthreads();

    // ---- scalar tail: one batch row per lane (lanes 0..15 of each wave) ----
    if (waveActive && lane < 16) {
        const int row = rowBase + lane;
        if (row < batch) {
            // tanh(pre) * pi/2
            float th0 = tanhf(sC[wave][lane][0] + preB[0]) * 1.57079632679489662f;
            float th1 = tanhf(sC[wave][lane][1] + preB[1]) * 1.57079632679489662f;
            float th2 = tanhf(sC[wave][lane][2] + preB[2]) * 1.57079632679489662f;

            // |000> after H on all three qubits: uniform amplitudes 8^-0.5
            float s[8];
            #pragma unroll
            for (int i = 0; i < 8; ++i) s[i] = 0.35355339059327373f;

            // RY on qubit q (q0 = MSB -> bit 2): [[c,-s],[s,c]]
            auto ry = [&](int bit, float theta) {
                float sn, c;
                __sincosf(theta * 0.5f, &sn, &c);
                #pragma unroll
                for (int i = 0; i < 8; ++i) {
                    if (!(i & bit)) {
                        int j = i | bit;
                        float a0 = s[i], a1 = s[j];
                        s[i] = c * a0 - sn * a1;
                        s[j] = sn * a0 + c * a1;
                    }
                }
            };

            // data-encoding RY layer (per-sample angles)
            ry(4, th0);  // qubit 0 -> bit 2
            ry(2, th1);  // qubit 1 -> bit 1
            ry(1, th2);  // qubit 2 -> bit 0

            // two entangling + trainable layers; uses q_params rows 1 and 2
            #pragma unroll
            for (int k = 0; k < 2; ++k) {
                // CNOT(0,1): ctrl bit2, tgt bit1 -> swap (4,6),(5,7)
                { float t = s[4]; s[4] = s[6]; s[6] = t; }
                { float t = s[5]; s[5] = s[7]; s[7] = t; }
                // CNOT(1,2): ctrl bit1, tgt bit0 -> swap (2,3),(6,7)
                { float t = s[2]; s[2] = s[3]; s[3] = t; }
                { float t = s[6]; s[6] = s[7]; s[7] = t; }
                ry(4, qp[3 * (k + 1) + 0]);
                ry(2, qp[3 * (k + 1) + 1]);
                ry(1, qp[3 * (k + 1) + 2]);
            }

            // probs @ ZSIGNS  (ZSIGNS[i][j] = 1 - 2*bit_{2-j}(i))
            float q0 = 0.0f, q1 = 0.0f, q2 = 0.0f;
            #pragma unroll
            for (int i = 0; i < 8; ++i) {
                float p = s[i] * s[i];
                q0 += (i & 4) ? -p : p;
                q1 += (i & 2) ? -p : p;
                q2 += (i & 1) ? -p : p;
            }

            // post linear: (3) -> (10)
            float* o = out + (size_t)row * NOUT;
            #pragma unroll
            for (int j = 0; j < NOUT; ++j)
                o[j] = q0 * postW[3 * j + 0] + q1 * postW[3 * j + 1]
                     + q2 * postW[3 * j + 2] + postB[j];
        }
    }
}

extern "C" void kernel_launch(void* const* d_in, const int* in_sizes, int n_in,
                              void* d_out, int out_size, void* d_ws, size_t ws_size,
                              hipStream_t stream) {
    const float* X     = (const float*)d_in[0];
    const float* preW  = (const float*)d_in[1];
    const float* preB  = (const float*)d_in[2];
    const float* qp    = (const float*)d_in[3];
    const float* postW = (const float*)d_in[4];
    const float* postB = (const float*)d_in[5];
    float* out = (float*)d_out;

    const int batch = in_sizes[0] / KDIM;               // 65536
    const int rowsPerBlock = 128;                       // 8 waves x 16 rows
    const int grid = (batch + rowsPerBlock - 1) / rowsPerBlock;

    qnet_fused_kernel<<<grid, 256, 0, stream>>>(X, preW, preB, qp, postW, postB,
                                                out, batch);
}